// IRNNlayer_635655160296
// MI455X (gfx1250) — compile-verified
//
#include <hip/hip_runtime.h>

#define EPS   1e-5f
#define CH    32
#define NSPAT (64*64*64)        // 262144 spatial positions
#define N1    (CH*NSPAT)        // 8388608 elements per 32-ch volume
#define NSL   (CH*64*64)        // 131072 elements per direction slice
#define HB_FLOATS (6*NSL)       // 786432

typedef float v2f __attribute__((ext_vector_type(2)));
typedef float v8f __attribute__((ext_vector_type(8)));

static __device__ __forceinline__ v8f wmma4(v2f a, v2f b, v8f c) {
  // D = A(16x4 f32) * B(4x16 f32) + C(16x16 f32), full fp32 precision
  return __builtin_amdgcn_wmma_f32_16x16x4_f32(false, a, false, b, (short)0, c,
                                               false, false);
}

static __device__ __forceinline__ int iclamp64(int v) {
  v = v < 0 ? 0 : v;
  return v > 63 ? 63 : v;
}

// ---------------------------------------------------------------- utilities
__global__ void k_zero(float* __restrict__ p, long n) {
  long i = (long)blockIdx.x * blockDim.x + threadIdx.x;
  long s = (long)gridDim.x * blockDim.x;
  for (; i < n; i += s) p[i] = 0.f;
}

// ------------------------------------------------- stage 1: conv3d + stats
// Implicit GEMM: M=32 (Ch), K=27 taps x 16 ch (tap-major), N=262144.
// Block = 8 waves, each wave one 16x16 tile; block covers one (z,y) x-row.
__global__ __launch_bounds__(256)
void k_conv3d(const float* __restrict__ x, const float* __restrict__ w,
              float* __restrict__ y1, float* __restrict__ stats) {
  __shared__ float As[32 * 432];           // permuted: [o][kidx*16 + c]
  __shared__ float rs[2];
  const int tid = threadIdx.x;
  for (int i = tid; i < 32 * 432; i += 256) {
    const int o = i / 432, rem = i % 432;
    const int c = rem / 27, kidx = rem % 27;
    As[o * 432 + kidx * 16 + c] = w[i];
  }
  if (tid == 0) { rs[0] = 0.f; rs[1] = 0.f; }
  __syncthreads();

  const int z = blockIdx.x >> 6, y = blockIdx.x & 63;
  const int wave = tid >> 5, lane = tid & 31;
  const int mi = wave & 1, ni = wave >> 1;      // 2 M-tiles x 4 N-tiles
  const int half = lane >> 4, ln = lane & 15;
  const int xx = ni * 16 + ln;
  const float* Arow = As + (mi * 16 + ln) * 432;

  v8f acc = {0.f,0.f,0.f,0.f,0.f,0.f,0.f,0.f};
#pragma unroll
  for (int kidx = 0; kidx < 27; ++kidx) {
    const int dz = kidx / 9 - 1, dy = (kidx % 9) / 3 - 1, dx = kidx % 3 - 1;
    const int zz = z + dz, yy = y + dy, xv = xx + dx;
    const bool inb = ((unsigned)zz < 64u) && ((unsigned)yy < 64u) &&
                     ((unsigned)xv < 64u);
    const float msk = inb ? 1.f : 0.f;
    const float* Xp = x + iclamp64(zz) * 4096 + iclamp64(yy) * 64 + iclamp64(xv);
#pragma unroll
    for (int c0 = 0; c0 < 16; c0 += 4) {
      v2f a, b;
      b.x = Xp[(c0 + half) << 18] * msk;        // always-valid load + mask
      b.y = Xp[(c0 + 2 + half) << 18] * msk;
      const int ka = kidx * 16 + c0 + 2 * half;
      a.x = Arow[ka]; a.y = Arow[ka + 1];
      acc = wmma4(a, b, acc);
    }
  }

  float ls = 0.f, lq = 0.f;
  const int base = z * 4096 + y * 64 + xx;
#pragma unroll
  for (int r = 0; r < 8; ++r) {
    const int m = mi * 16 + r + 8 * half;       // C/D: M = r + 8*(lane/16)
    const float v = acc[r];
    y1[(m << 18) + base] = v;
    ls += v; lq += v * v;
  }
  atomicAdd(&rs[0], ls); atomicAdd(&rs[1], lq);
  __syncthreads();
  if (tid == 0) { atomicAdd(&stats[0], rs[0]); atomicAdd(&stats[1], rs[1]); }
}

__global__ __launch_bounds__(256)
void k_norm1(const float* __restrict__ y1, const float* __restrict__ stats,
             const float* __restrict__ g, const float* __restrict__ b,
             float* __restrict__ xh) {
  const float invN = 1.f / (float)N1;
  const float mu = stats[0] * invN;
  const float var = stats[1] * invN - mu * mu;
  const float rsig = rsqrtf(var + EPS);
  const int i = blockIdx.x * 256 + threadIdx.x;
  const int c = i >> 18;
  const float v = (y1[i] - mu) * rsig * g[c] + b[c];
  xh[i] = fmaxf(v, 0.f);
}

// ---------------------------------------- stage 2: 6 persistent IRNN scans
// One block per direction; 32 waves; per step:
//   conv2d GEMM (M=32, K=9 taps x 32 ch tap-major, N=4096) -> stats reduce
//   -> normalize+relu update -> w_r 32x32 projection atomically into y.
__global__ __launch_bounds__(1024)
void k_scan(const float* __restrict__ xh,
            const float* __restrict__ wh, const float* __restrict__ gh,
            const float* __restrict__ bh, const float* __restrict__ wr,
            float* __restrict__ hbuf, float* __restrict__ zbuf,
            float* __restrict__ yacc) {
  const int d = blockIdx.x;
  const int tid = threadIdx.x;
  __shared__ float Wh[32 * 288];     // permuted: [m][kidx*32 + c]
  __shared__ float Wr[32 * 32];      // this direction's w_r column block
  __shared__ float Gm[32], Bt[32];
  __shared__ float rs[2];
  for (int i = tid; i < 32 * 288; i += 1024) {
    const int m = i / 288, rem = i % 288;
    const int c = rem / 9, kidx = rem % 9;
    Wh[m * 288 + kidx * 32 + c] = wh[d * 9216 + i];
  }
  Wr[tid & 1023] = wr[((tid & 1023) >> 5) * 192 + d * 32 + (tid & 31)];
  if (tid < 32) { Gm[tid] = gh[d * 32 + tid]; Bt[tid] = bh[d * 32 + tid]; }
  __syncthreads();

  float* H = hbuf + d * NSL;         // hidden state (zero-initialized)
  float* Z = zbuf + d * NSL;         // raw conv output scratch
  const int wave = tid >> 5, lane = tid & 31;
  const int half = lane >> 4, ln = lane & 15;
  const float invN = 1.f / (float)NSL;
  const float* A0 = Wh + ln * 288;
  const float* A1 = Wh + (16 + ln) * 288;

  // direction geometry: spatial offset = base + u*cu + v*cv
  int cu, cv;
  switch (d >> 1) {
    case 0:  cu = 64;   cv = 1;    break;   // seq = D, slice (H,W)
    case 1:  cu = 4096; cv = 1;    break;   // seq = H, slice (D,W)
    default: cu = 64;   cv = 4096; break;   // seq = W, slice (H,D)
  }
  const int seq_mul = (d >> 1) == 0 ? 4096 : ((d >> 1) == 1 ? 64 : 1);
  const bool rev = (d & 1) == 0;

  for (int t = 0; t < 64; ++t) {
    const int rbase = (rev ? 63 - t : t) * seq_mul;   // input slice position
    const int wbase = t * seq_mul;                    // output position
    float ls = 0.f, lq = 0.f;
    // ---- phase 1: z = conv2d(h, Wh); each wave: 2 M-tiles x 8 N-tiles
    for (int j = 0; j < 8; ++j) {
      const int p0 = (wave * 8 + j) * 16 + ln;
      const int u = p0 >> 6, v = p0 & 63;
      v8f acc0 = {0.f,0.f,0.f,0.f,0.f,0.f,0.f,0.f};
      v8f acc1 = {0.f,0.f,0.f,0.f,0.f,0.f,0.f,0.f};
#pragma unroll
      for (int kidx = 0; kidx < 9; ++kidx) {
        const int dy = kidx / 3 - 1, dx = kidx % 3 - 1;
        const int uu = u + dy, vv = v + dx;
        const bool inb = ((unsigned)uu < 64u) && ((unsigned)vv < 64u);
        const float msk = inb ? 1.f : 0.f;
        const float* Hp = H + iclamp64(uu) * 64 + iclamp64(vv);
#pragma unroll
        for (int c0 = 0; c0 < 32; c0 += 4) {
          v2f a0, a1, b;
          b.x = Hp[(c0 + half) << 12] * msk;
          b.y = Hp[(c0 + 2 + half) << 12] * msk;
          const int ka = kidx * 32 + c0 + 2 * half;
          a0.x = A0[ka]; a0.y = A0[ka + 1];
          a1.x = A1[ka]; a1.y = A1[ka + 1];
          acc0 = wmma4(a0, b, acc0);
          acc1 = wmma4(a1, b, acc1);
        }
      }
#pragma unroll
      for (int r = 0; r < 8; ++r) {
        const int m = r + 8 * half;
        const float v0 = acc0[r], v1 = acc1[r];
        Z[(m << 12) + p0] = v0;
        Z[((16 + m) << 12) + p0] = v1;
        ls += v0 + v1; lq += v0 * v0 + v1 * v1;
      }
    }
    // ---- phase 2: slice-wide mean/var
    if (tid == 0) { rs[0] = 0.f; rs[1] = 0.f; }
    __syncthreads();
    atomicAdd(&rs[0], ls); atomicAdd(&rs[1], lq);
    __syncthreads();
    const float mu = rs[0] * invN;
    const float var = rs[1] * invN - mu * mu;
    const float rsig = rsqrtf(var + EPS);
    // ---- phase 3: h = relu(x_t + groupnorm(z))
    for (int i = tid; i < NSL; i += 1024) {
      const int c = i >> 12, p = i & 4095;
      const int u = p >> 6, v = p & 63;
      const float zz = Z[i];
      const float xt = xh[(c << 18) + rbase + u * cu + v * cv];
      const float hv = xt + (zz - mu) * rsig * Gm[c] + Bt[c];
      H[i] = fmaxf(hv, 0.f);
    }
    __syncthreads();
    // ---- phase 4: y += w_r_d @ h  (M=32, K=32, N=4096), atomic accumulate
    for (int j = 0; j < 8; ++j) {
      const int p0 = (wave * 8 + j) * 16 + ln;
      v8f acc0 = {0.f,0.f,0.f,0.f,0.f,0.f,0.f,0.f};
      v8f acc1 = {0.f,0.f,0.f,0.f,0.f,0.f,0.f,0.f};
      const float* Hp = H + p0;
#pragma unroll
      for (int k0 = 0; k0 < 32; k0 += 4) {
        v2f a0, a1, b;
        b.x = Hp[(k0 + half) << 12];
        b.y = Hp[(k0 + 2 + half) << 12];
        const int ka = k0 + 2 * half;
        a0.x = Wr[ln * 32 + ka];        a0.y = Wr[ln * 32 + ka + 1];
        a1.x = Wr[(16 + ln) * 32 + ka]; a1.y = Wr[(16 + ln) * 32 + ka + 1];
        acc0 = wmma4(a0, b, acc0);
        acc1 = wmma4(a1, b, acc1);
      }
      const int u = p0 >> 6, v = p0 & 63;
      const int woff = wbase + u * cu + v * cv;
#pragma unroll
      for (int r = 0; r < 8; ++r) {
        const int m = r + 8 * half;
        atomicAdd(&yacc[(m << 18) + woff], acc0[r]);
        atomicAdd(&yacc[((16 + m) << 18) + woff], acc1[r]);
      }
    }
  }
}

// ------------------------------------------------- stage 3: final groupnorm
__global__ __launch_bounds__(256)
void k_ystats(const float* __restrict__ y, float* __restrict__ stats) {
  __shared__ float rs[2];
  float ls = 0.f, lq = 0.f;
  for (long i = (long)blockIdx.x * blockDim.x + threadIdx.x; i < (long)N1;
       i += (long)gridDim.x * blockDim.x) {
    const float v = y[i];
    ls += v; lq += v * v;
  }
  if (threadIdx.x == 0) { rs[0] = 0.f; rs[1] = 0.f; }
  __syncthreads();
  atomicAdd(&rs[0], ls); atomicAdd(&rs[1], lq);
  __syncthreads();
  if (threadIdx.x == 0) { atomicAdd(&stats[2], rs[0]); atomicAdd(&stats[3], rs[1]); }
}

__global__ __launch_bounds__(256)
void k_final(const float* __restrict__ y, const float* __restrict__ stats,
             const float* __restrict__ g, const float* __restrict__ b,
             float* __restrict__ out) {
  const float invN = 1.f / (float)N1;
  const float mu = stats[2] * invN;
  const float var = stats[3] * invN - mu * mu;
  const float rsig = rsqrtf(var + EPS);
  const int i = blockIdx.x * 256 + threadIdx.x;
  const int c = i >> 18;
  const float v = (y[i] - mu) * rsig * g[c] + b[c];
  out[i] = fmaxf(v, 0.f);
}

// ------------------------------------------------------------------- host
extern "C" void kernel_launch(void* const* d_in, const int* in_sizes, int n_in,
                              void* d_out, int out_size, void* d_ws, size_t ws_size,
                              hipStream_t stream) {
  const float* x   = (const float*)d_in[0];
  const float* w_x = (const float*)d_in[1];
  const float* g_x = (const float*)d_in[2];
  const float* b_x = (const float*)d_in[3];
  const float* w_h = (const float*)d_in[4];
  const float* g_h = (const float*)d_in[5];
  const float* b_h = (const float*)d_in[6];
  const float* w_r = (const float*)d_in[7];
  const float* g_r = (const float*)d_in[8];
  const float* b_r = (const float*)d_in[9];
  float* out = (float*)d_out;

  // workspace layout (floats): ~73 MB total, fits the 192 MB L2
  float* ws    = (float*)d_ws;
  float* bufA  = ws;                          // conv3d raw out, reused as y accum
  float* xh    = ws + (size_t)N1;             // x_hidden
  float* hbuf  = ws + 2 * (size_t)N1;         // 6 hidden-state slices
  float* zbuf  = hbuf + HB_FLOATS;            // 6 conv scratch slices
  float* stats = zbuf + HB_FLOATS;            // [0..1] stage1, [2..3] final

  k_zero<<<2048, 256, 0, stream>>>(hbuf, (long)(2 * HB_FLOATS + 64));
  k_conv3d<<<4096, 256, 0, stream>>>(x, w_x, bufA, stats);
  k_norm1<<<N1 / 256, 256, 0, stream>>>(bufA, stats, g_x, b_x, xh);
  k_zero<<<4096, 256, 0, stream>>>(bufA, (long)N1);   // y accumulator
  k_scan<<<6, 1024, 0, stream>>>(xh, w_h, g_h, b_h, w_r, hbuf, zbuf, bufA);
  k_ystats<<<1024, 256, 0, stream>>>(bufA, stats);
  k_final<<<N1 / 256, 256, 0, stream>>>(bufA, stats, g_r, b_r, out);
  (void)in_sizes; (void)n_in; (void)out_size; (void)ws_size;
}